// LIFSpike_84542136254876
// MI455X (gfx1250) — compile-verified
//
#include <hip/hip_runtime.h>

// LIF spiking-neuron scan, T=16 leading time axis.
//   u_t = u_{t-1} - Vth*o_{t-1} + x_t ;  o_t = (u_t > Vth) ? 1 : 0
// Memory-bound: 512MB streamed @ 23.3 TB/s -> ~22us floor. No matmul -> no WMMA.
// CDNA5 path: async global->LDS staging (ASYNCcnt) + pipelined s_wait_asynccnt.

typedef __attribute__((ext_vector_type(4))) float v4f;
typedef __attribute__((ext_vector_type(4))) int   v4i;

#define STEPS 16
#define VTH   1.0f
#define TPB   256

#define AS1 __attribute__((address_space(1)))
#define AS3 __attribute__((address_space(3)))

// ---- feature detection (async builtins are probe-confirmed present on
//      gfx1250 device pass; host pass takes the fallback branch) ----
#if defined(__gfx1250__) &&                                         \
    __has_builtin(__builtin_amdgcn_global_load_async_to_lds_b128) && \
    __has_builtin(__builtin_amdgcn_s_wait_asynccnt)
#define USE_ASYNC_LDS 1
#else
#define USE_ASYNC_LDS 0
#endif

__device__ __forceinline__ v4f spike4(v4f u) {
  v4f o;
  o.x = u.x > VTH ? 1.0f : 0.0f;
  o.y = u.y > VTH ? 1.0f : 0.0f;
  o.z = u.z > VTH ? 1.0f : 0.0f;
  o.w = u.w > VTH ? 1.0f : 0.0f;
  return o;
}

#if USE_ASYNC_LDS
// Template-unrolled time loop: s_wait_asynccnt needs an integer *constant*,
// so the per-step wait count (STEPS-1-t) must be a template parameter.
// Async loads retire in order -> step t only needs asynccnt <= 15-t.
template <int t>
__device__ __forceinline__ void lif_steps(const v4f (*tile)[TPB], v4f& u, v4f& o,
                                          v4f* __restrict__ out, int n4, int i,
                                          int tid) {
  if constexpr (t < STEPS) {
    __builtin_amdgcn_s_wait_asynccnt(STEPS - 1 - t);
    asm volatile("" ::: "memory");  // keep the LDS read below the wait
    v4f xt = tile[t][tid];
    u = u - o + xt;                 // Vth == 1
    o = spike4(u);
    __builtin_nontemporal_store(o, &out[t * n4 + i]);
    lif_steps<t + 1>(tile, u, o, out, n4, i, tid);
  }
}
#endif

__global__ __launch_bounds__(TPB) void lif_scan_kernel(const v4f* __restrict__ x,
                                                       v4f* __restrict__ out,
                                                       int n4) {
  const int tid = threadIdx.x;
  const int i   = blockIdx.x * TPB + tid;
  if (i >= n4) return;

#if USE_ASYNC_LDS
  // Stage all 16 time-planes of this block's tile into LDS via the CDNA5
  // async DMA path. 16 steps * 256 thr * 16B = 64KB LDS (of 320KB/WGP).
  // Each thread reads back only bytes its own wave's async ops wrote, and
  // async loads retire in order -> per-step wait, no barrier needed.
  __shared__ v4f tile[STEPS][TPB];
#pragma unroll
  for (int t = 0; t < STEPS; ++t) {
    __builtin_amdgcn_global_load_async_to_lds_b128(
        (AS1 v4i*)(&x[t * n4 + i]),
        (AS3 v4i*)(&tile[t][tid]),
        /*offset=*/0, /*cpol=*/0);
  }

  v4f u = (v4f)0.0f, o = (v4f)0.0f;
  lif_steps<0>(tile, u, o, out, n4, i, tid);
#else
  // Fallback: preload all 16 planes into registers (16 outstanding b128
  // loads per thread), then run the recurrence.
  v4f xs[STEPS];
#pragma unroll
  for (int t = 0; t < STEPS; ++t)
    xs[t] = __builtin_nontemporal_load(&x[t * n4 + i]);

  v4f u = (v4f)0.0f, o = (v4f)0.0f;
#pragma unroll
  for (int t = 0; t < STEPS; ++t) {
    u = u - o + xs[t];              // Vth == 1
    o = spike4(u);
    __builtin_nontemporal_store(o, &out[t * n4 + i]);
  }
#endif
}

extern "C" void kernel_launch(void* const* d_in, const int* in_sizes, int n_in,
                              void* d_out, int out_size, void* d_ws, size_t ws_size,
                              hipStream_t stream) {
  const float* x = (const float*)d_in[0];
  float* out     = (float*)d_out;

  const int total = in_sizes[0];      // T*B*C*H*W = 67,108,864
  const int N     = total / STEPS;    // spatial elements = 4,194,304
  const int n4    = N / 4;            // float4 groups   = 1,048,576

  const int blocks = (n4 + TPB - 1) / TPB;  // 4096
  lif_scan_kernel<<<blocks, TPB, 0, stream>>>((const v4f*)x, (v4f*)out, n4);
}